// Model_57294863729186
// MI455X (gfx1250) — compile-verified
//
#include <hip/hip_runtime.h>

typedef __attribute__((ext_vector_type(2))) float v2f;
typedef __attribute__((ext_vector_type(8))) float v8f;

#define ALPHA_ 0.2f
#define NEGV_  -9e15f

constexpr int B_ = 2, N_ = 4096, D_ = 128, H_ = 4, HID_ = 32, EN_ = 16;
constexpr int NTILES_ = N_ / 64;   // 64-column tiles per row

// ---------------- Pack binary adjacency into 64-bit masks: mask[b,row,tile]
__global__ __launch_bounds__(256) void adj_mask_k(
    const float* __restrict__ adj,            // [B,N,N]
    unsigned long long* __restrict__ mask)    // [B,N,N/64]
{
    int w = (blockIdx.x * 256 + threadIdx.x) >> 5;   // wave id = b*N + row
    int lane = threadIdx.x & 31;
    int row = w % N_, b = w / N_;
    const float* ar = adj + ((size_t)b * N_ + row) * N_;
    unsigned long long* mr = mask + ((size_t)b * N_ + row) * NTILES_;
    for (int jt = 0; jt < NTILES_; ++jt) {
        float a0 = ar[jt * 64 + lane];
        float a1 = ar[jt * 64 + 32 + lane];
        unsigned long long b0 = __ballot(a0 != 0.f);
        unsigned long long b1 = __ballot(a1 != 0.f);
        if (lane == 0) mr[jt] = (b0 & 0xffffffffull) | (b1 << 32);
    }
}

// ---------------- Projection: wh = x @ W_h ; wh1 = wh@a[:HID]; wh2 = wh@a[HID:]
__global__ __launch_bounds__(256) void proj_heads_k(
    const float* __restrict__ x, const float* __restrict__ W,
    const float* __restrict__ a,
    float* __restrict__ wh, float* __restrict__ wh1, float* __restrict__ wh2)
{
    int t = blockIdx.x * 256 + threadIdx.x;
    int n = t % N_;
    int h = (t / N_) % H_;
    int b = t / (N_ * H_);
    const float* xr = x + ((size_t)b * N_ + n) * D_;
    const float* Wh = W + (size_t)h * D_ * HID_;
    float acc[HID_];
#pragma unroll
    for (int c = 0; c < HID_; ++c) acc[c] = 0.f;
    for (int k = 0; k < D_; ++k) {
        float xv = xr[k];
        const float* wr = Wh + k * HID_;
#pragma unroll
        for (int c = 0; c < HID_; ++c) acc[c] = fmaf(xv, wr[c], acc[c]);
    }
    const float* ah = a + h * 2 * HID_;
    float s1 = 0.f, s2 = 0.f;
    float* whr = wh + ((size_t)(b * H_ + h) * N_ + n) * HID_;
#pragma unroll
    for (int c = 0; c < HID_; ++c) {
        whr[c] = acc[c];
        s1 = fmaf(acc[c], ah[c], s1);
        s2 = fmaf(acc[c], ah[HID_ + c], s2);
    }
    wh1[(size_t)(b * H_ + h) * N_ + n] = s1;
    wh2[(size_t)(b * H_ + h) * N_ + n] = s2;
}

__global__ __launch_bounds__(256) void proj_last_k(
    const float* __restrict__ xc, const float* __restrict__ W,
    const float* __restrict__ a,
    float* __restrict__ wh, float* __restrict__ wh1, float* __restrict__ wh2)
{
    int t = blockIdx.x * 256 + threadIdx.x;
    int n = t % N_;
    int b = t / N_;
    const float* xr = xc + ((size_t)b * N_ + n) * (H_ * HID_);
    float acc[EN_];
#pragma unroll
    for (int c = 0; c < EN_; ++c) acc[c] = 0.f;
    for (int k = 0; k < H_ * HID_; ++k) {
        float xv = xr[k];
        const float* wr = W + k * EN_;
#pragma unroll
        for (int c = 0; c < EN_; ++c) acc[c] = fmaf(xv, wr[c], acc[c]);
    }
    float s1 = 0.f, s2 = 0.f;
    float* whr = wh + ((size_t)b * N_ + n) * EN_;
#pragma unroll
    for (int c = 0; c < EN_; ++c) {
        whr[c] = acc[c];
        s1 = fmaf(acc[c], a[c], s1);
        s2 = fmaf(acc[c], a[EN_ + c], s2);
    }
    wh1[(size_t)b * N_ + n] = s1;
    wh2[(size_t)b * N_ + n] = s2;
}

// ---------------- Fused masked-softmax attention + P@V via WMMA f32 16x16x4
// Block = 128 threads = 4 waves; wave handles 16 rows. 64-col tiles streamed with
// double-buffered async-to-LDS V tiles; adjacency consumed as 64-bit masks.
template <int OUT>
__global__ __launch_bounds__(128) void gat_attn_k(
    const unsigned long long* __restrict__ mask,  // [B,N,N/64]
    const float* __restrict__ wh,                 // [B*gridDim.y, N, OUT]
    const float* __restrict__ wh1,
    const float* __restrict__ wh2,
    float* __restrict__ out, int outRS)
{
    constexpr int NT = OUT / 16;      // 16-wide C tiles
    constexpr int VS = OUT + 4;       // LDS float stride for V tile
    const int h = blockIdx.y, b = blockIdx.z;
    const int rowbase = blockIdx.x * 64;
    const int wave = threadIdx.x >> 5;
    const int lane = threadIdx.x & 31;
    const int rlo  = lane & 15;
    const int half = lane >> 4;
    const int wrow = rowbase + wave * 16;

    __shared__ alignas(16) float vT[2][64 * VS];
    __shared__ alignas(16) float w2T[2][64];

    const size_t lOff = (size_t)(b * gridDim.y + h) * N_;
    const float* whB  = wh + lOff * OUT;
    const float  w1r  = wh1[lOff + wrow + rlo];
    const float* w2B  = wh2 + lOff;
    const unsigned long long* mB =
        mask + ((size_t)b * N_ + wrow + rlo) * NTILES_;

    float m = -INFINITY, l = 0.f;
    v8f acc[NT];
#pragma unroll
    for (int t = 0; t < NT; ++t)
#pragma unroll
        for (int r = 0; r < 8; ++r) acc[t][r] = 0.f;

    // async issue of V tile + wh2 tile for tile `ti` into buffer `buf`
    auto issue = [&](int ti, int buf) {
        const float* gv = whB + (size_t)ti * 64 * OUT;
        unsigned vbase = (unsigned)(uintptr_t)(&vT[buf][0]);
#pragma unroll
        for (int k = 0; k < (64 * OUT / 4) / 128; ++k) {
            int i = threadIdx.x + k * 128;
            int r = i / (OUT / 4), c4 = i % (OUT / 4);
            unsigned ldst = vbase + (unsigned)((r * VS + c4 * 4) * 4);
            unsigned long long ga = (unsigned long long)(uintptr_t)(gv + r * OUT + c4 * 4);
            asm volatile("global_load_async_to_lds_b128 %0, %1, off"
                         :: "v"(ldst), "v"(ga) : "memory");
        }
        if (threadIdx.x < 16) {
            unsigned ldst = (unsigned)(uintptr_t)(&w2T[buf][0]) + threadIdx.x * 16u;
            unsigned long long ga =
                (unsigned long long)(uintptr_t)(w2B + ti * 64 + threadIdx.x * 4);
            asm volatile("global_load_async_to_lds_b128 %0, %1, off"
                         :: "v"(ldst), "v"(ga) : "memory");
        }
    };

    issue(0, 0);
    for (int ti = 0; ti < NTILES_; ++ti) {
        const int cur = ti & 1;
        asm volatile("s_wait_asynccnt 0x0" ::: "memory");
        __syncthreads();
        if (ti + 1 < NTILES_) issue(ti + 1, cur ^ 1);

        unsigned long long mk = mB[ti];   // adjacency bits for this row's 64 cols

        // ---- masked leaky-relu scores directly in WMMA A-fragment layout
        v2f p[16];
        float tmax = -INFINITY;
#pragma unroll
        for (int kf = 0; kf < 16; ++kf) {
            int c0 = kf * 4 + half * 2;
            float e0 = w1r + w2T[cur][c0];
            float e1 = w1r + w2T[cur][c0 + 1];
            e0 = e0 > 0.f ? e0 : ALPHA_ * e0;
            e1 = e1 > 0.f ? e1 : ALPHA_ * e1;
            unsigned bits = (unsigned)(mk >> c0) & 3u;
            float s0 = ((bits & 1u) && e0 != 0.f) ? e0 : NEGV_;
            float s1 = ((bits & 2u) && e1 != 0.f) ? e1 : NEGV_;
            p[kf].x = s0; p[kf].y = s1;
            tmax = fmaxf(tmax, fmaxf(s0, s1));
        }
        tmax = fmaxf(tmax, __shfl_xor(tmax, 16));
        float mn = fmaxf(m, tmax);
        float scale = __expf(m - mn);
        float tsum = 0.f;
#pragma unroll
        for (int kf = 0; kf < 16; ++kf) {
            float p0 = __expf(p[kf].x - mn);
            float p1 = __expf(p[kf].y - mn);
            p[kf].x = p0; p[kf].y = p1;
            tsum += p0 + p1;
        }
        tsum += __shfl_xor(tsum, 16);
        l = l * scale + tsum;
        m = mn;
#pragma unroll
        for (int r = 0; r < 8; ++r) {
            float sc = __shfl(scale, r + (half << 3));
#pragma unroll
            for (int t = 0; t < NT; ++t) acc[t][r] *= sc;
        }
        // ---- P @ V : 16 K-fragments of 4, NT column tiles
#pragma unroll
        for (int kf = 0; kf < 16; ++kf) {
            int kr = kf * 4 + half * 2;
#pragma unroll
            for (int t = 0; t < NT; ++t) {
                v2f bf;
                bf.x = vT[cur][(kr + 0) * VS + t * 16 + rlo];
                bf.y = vT[cur][(kr + 1) * VS + t * 16 + rlo];
                acc[t] = __builtin_amdgcn_wmma_f32_16x16x4_f32(
                    false, p[kf], false, bf, (short)0, acc[t], false, false);
            }
        }
    }

    // ---- normalize, ELU, store
    float inv = 1.f / l;
    const int colOff = h * OUT;
#pragma unroll
    for (int r = 0; r < 8; ++r) {
        float iv = __shfl(inv, r + (half << 3));
        int row = wrow + r + half * 8;
#pragma unroll
        for (int t = 0; t < NT; ++t) {
            float v = acc[t][r] * iv;
            v = v > 0.f ? v : (__expf(v) - 1.f);
            out[((size_t)b * N_ + row) * outRS + colOff + t * 16 + rlo] = v;
        }
    }
}

extern "C" void kernel_launch(void* const* d_in, const int* in_sizes, int n_in,
                              void* d_out, int out_size, void* d_ws, size_t ws_size,
                              hipStream_t stream) {
    const float* fea   = (const float*)d_in[0];
    const float* adj   = (const float*)d_in[1];
    const float* Whead = (const float*)d_in[2];
    const float* ahead = (const float*)d_in[3];
    const float* Wlast = (const float*)d_in[4];
    const float* alast = (const float*)d_in[5];
    float* outp = (float*)d_out;

    unsigned long long* maskp = (unsigned long long*)d_ws;        // B*N*(N/64) u64 = 4MB
    float* ws = (float*)(maskp + (size_t)B_ * N_ * NTILES_);
    float* wh_h  = ws;
    float* wh1_h = wh_h  + (size_t)B_ * H_ * N_ * HID_;
    float* wh2_h = wh1_h + (size_t)B_ * H_ * N_;
    float* xcat  = wh2_h + (size_t)B_ * H_ * N_;
    float* wh_l  = xcat  + (size_t)B_ * N_ * H_ * HID_;
    float* wh1_l = wh_l  + (size_t)B_ * N_ * EN_;
    float* wh2_l = wh1_l + (size_t)B_ * N_;

    // 0) pack adjacency into bit masks (one fp32 pass over adj, ever)
    adj_mask_k<<<(B_ * N_) / 8, 256, 0, stream>>>(adj, maskp);

    // 1) per-head projections
    proj_heads_k<<<(B_ * H_ * N_) / 256, 256, 0, stream>>>(
        fea, Whead, ahead, wh_h, wh1_h, wh2_h);

    // 2) fused masked softmax + P@V for 4 heads -> xcat (ELU, concat layout)
    gat_attn_k<HID_><<<dim3(N_ / 64, H_, B_), 128, 0, stream>>>(
        maskp, wh_h, wh1_h, wh2_h, xcat, H_ * HID_);

    // 3) last projection
    proj_last_k<<<(B_ * N_) / 256, 256, 0, stream>>>(
        xcat, Wlast, alast, wh_l, wh1_l, wh2_l);

    // 4) last fused attention layer -> output (ELU)
    gat_attn_k<EN_><<<dim3(N_ / 64, 1, B_), 128, 0, stream>>>(
        maskp, wh_l, wh1_l, wh2_l, outp, EN_);
}